// means_cd_loss_7249904795879
// MI455X (gfx1250) — compile-verified
//
#include <hip/hip_runtime.h>
#include <math.h>

typedef __attribute__((ext_vector_type(2))) float v2f;
typedef __attribute__((ext_vector_type(8))) float v8f;

#define WAVES           8      // waves per block (256 threads, wave32)
#define STRIPE          512    // gt points per LDS stripe
#define CHUNK           16     // gt points per WMMA tile
#define MEANS_PER_WAVE  32     // two 16-wide mean tiles per wave
#define MEANS_PER_BLOCK 256
#define SPLITS          2      // gt range split per mean-set (grid width)

struct F3 { float x, y, z; };

// ---------------------------------------------------------------------------
// Init: both min arrays (d1: B*N, d2: B*M, contiguous) to +inf bits
// ---------------------------------------------------------------------------
__global__ void cd_init(unsigned int* mins, int total) {
    int i = blockIdx.x * blockDim.x + threadIdx.x;
    if (i < total) mins[i] = 0x7F800000u; // +inf
}

__device__ __forceinline__ float min8(const v8f& c) {
    float a = fminf(fminf(c[0], c[1]), fminf(c[2], c[3]));
    float b = fminf(fminf(c[4], c[5]), fminf(c[6], c[7]));
    return fminf(a, b);
}

// ---------------------------------------------------------------------------
// Main: block = 256 means of one batch, sweeping M/SPLITS gt points.
// Per 16-gt chunk, four V_WMMA_F32_16X16X4_F32 (two mean tiles x two
// distance directions), with the ||.||^2 terms folded into the K=4 slot:
//   A-row = (-2px,-2py,-2pz,||p||^2), B-col = (qx,qy,qz,1)
//   -> D[p][q] = ||p||^2 - 2 p.q  (other norm added after the min)
// gt chunk is prefetched one iteration ahead; both lane-halves write their
// partial d2 min rows to LDS unconditionally (no shuffles/branches in loop).
// ---------------------------------------------------------------------------
__global__ __launch_bounds__(256) void cd_main(const float* __restrict__ means,
                                               const float* __restrict__ gt,
                                               unsigned int* __restrict__ d1min,
                                               unsigned int* __restrict__ d2min,
                                               int B, int N, int M)
{
    __shared__ float lds[2 * WAVES][STRIPE]; // 32 KB

    const int blocksPerBatch = (N / MEANS_PER_BLOCK) * SPLITS;
    int r = blockIdx.x;
    const int b    = r / blocksPerBatch;  r %= blocksPerBatch;
    const int nb   = r / SPLITS;
    const int half = r % SPLITS;
    const int Mh   = M / SPLITS;
    const int goff = half * Mh;

    const int  wave = threadIdx.x >> 5;
    const int  lane = threadIdx.x & 31;
    const int  l15  = lane & 15;
    const bool hi   = lane >= 16;
    const int  row  = wave * 2 + (hi ? 1 : 0);

    // ---- this wave's 32 means (two 16-wide tiles), fixed for whole sweep ----
    const int mbase = nb * MEANS_PER_BLOCK + wave * MEANS_PER_WAVE;
    const F3 ma = *(const F3*)(means + ((size_t)b * N + mbase + l15) * 3);
    const F3 mb = *(const F3*)(means + ((size_t)b * N + mbase + 16 + l15) * 3);
    const float x2a = ma.x * ma.x + ma.y * ma.y + ma.z * ma.z;
    const float x2b = mb.x * mb.x + mb.y * mb.y + mb.z * mb.z;

    v2f b1a, b1b, a2a, a2b;
    if (!hi) {
        b1a.x = ma.x;  b1a.y = ma.y;  a2a.x = -2.f * ma.x; a2a.y = -2.f * ma.y;
        b1b.x = mb.x;  b1b.y = mb.y;  a2b.x = -2.f * mb.x; a2b.y = -2.f * mb.y;
    } else {
        b1a.x = ma.z;  b1a.y = 1.0f;  a2a.x = -2.f * ma.z; a2a.y = x2a;
        b1b.x = mb.z;  b1b.y = 1.0f;  a2b.x = -2.f * mb.z; a2b.y = x2b;
    }

    float acc1a = 3.0e38f, acc1b = 3.0e38f;
    const v8f zero = {};

    // ---- depth-1 prefetch of the gt chunk ----
    const float* gb = gt + ((size_t)b * M + goff) * 3;
    F3 g = *(const F3*)(gb + (size_t)l15 * 3);

    for (int s = 0; s < Mh; s += STRIPE) {
        for (int c = 0; c < STRIPE; c += CHUNK) {
            int nidx = s + c + CHUNK;
            nidx = (nidx < Mh) ? nidx : 0;
            F3 gn = *(const F3*)(gb + (size_t)(nidx + l15) * 3); // prefetch next

            const float y2 = g.x * g.x + g.y * g.y + g.z * g.z;
            v2f a1, b2;
            if (!hi) { a1.x = -2.f * g.x; a1.y = -2.f * g.y; b2.x = g.x; b2.y = g.y;  }
            else     { a1.x = -2.f * g.z; a1.y = y2;         b2.x = g.z; b2.y = 1.0f; }

            v8f c1a = __builtin_amdgcn_wmma_f32_16x16x4_f32(
                false, a1, false, b1a, (short)0, zero, false, false);
            v8f c1b = __builtin_amdgcn_wmma_f32_16x16x4_f32(
                false, a1, false, b1b, (short)0, zero, false, false);
            v8f c2a = __builtin_amdgcn_wmma_f32_16x16x4_f32(
                false, a2a, false, b2, (short)0, zero, false, false);
            v8f c2b = __builtin_amdgcn_wmma_f32_16x16x4_f32(
                false, a2b, false, b2, (short)0, zero, false, false);

            acc1a = fminf(acc1a, min8(c1a));               // d1: min over gt rows
            acc1b = fminf(acc1b, min8(c1b));
            // d2: min over this wave's 32 means for this lane's gt column;
            // both lane-halves store their own row (merged in stripe reduce)
            lds[row][c + l15] = fminf(min8(c2a), min8(c2b)) + y2;

            g = gn;
        }
        __syncthreads();
        // cross-wave (and cross-half) min, one atomicMin per gt per block
        for (int gi = threadIdx.x; gi < STRIPE; gi += 256) {
            float m = lds[0][gi];
#pragma unroll
            for (int w = 1; w < 2 * WAVES; ++w) m = fminf(m, lds[w][gi]);
            m = fmaxf(m, 1e-9f); // clamp -> non-negative -> uint order == float order
            atomicMin(&d2min[(size_t)b * M + (size_t)(goff + s + gi)],
                      __float_as_uint(m));
        }
        __syncthreads();
    }

    // ---- finalize d1 partials for this wave's 32 means ----
    acc1a = fminf(acc1a, __shfl_xor(acc1a, 16, 32)); // merge gt-row halves
    acc1b = fminf(acc1b, __shfl_xor(acc1b, 16, 32));
    if (!hi) {
        unsigned int* base = d1min + (size_t)b * N + mbase;
        atomicMin(base + l15,      __float_as_uint(fmaxf(acc1a + x2a, 1e-9f)));
        atomicMin(base + 16 + l15, __float_as_uint(fmaxf(acc1b + x2b, 1e-9f)));
    }
}

// ---------------------------------------------------------------------------
// Final: sum sqrt of both clamped min arrays, combine into the scalar
// ---------------------------------------------------------------------------
__global__ __launch_bounds__(256) void cd_final(const unsigned int* __restrict__ d1min,
                                                const unsigned int* __restrict__ d2min,
                                                float* out, int B, int N, int M)
{
    __shared__ float sbuf[256];
    const int t1 = B * N, t2 = B * M;
    float s1 = 0.0f, s2 = 0.0f;
    for (int i = threadIdx.x; i < t1; i += 256) s1 += sqrtf(__uint_as_float(d1min[i]));
    for (int i = threadIdx.x; i < t2; i += 256) s2 += sqrtf(__uint_as_float(d2min[i]));
    sbuf[threadIdx.x] = s1 / (float)t1 + s2 / (float)t2;
    __syncthreads();
#pragma unroll
    for (int off = 128; off >= 1; off >>= 1) {
        if ((int)threadIdx.x < off) sbuf[threadIdx.x] += sbuf[threadIdx.x + off];
        __syncthreads();
    }
    if (threadIdx.x == 0) out[0] = sbuf[0] * 500.0f; // *0.5 * 1000
}

// ---------------------------------------------------------------------------
extern "C" void kernel_launch(void* const* d_in, const int* in_sizes, int n_in,
                              void* d_out, int out_size, void* d_ws, size_t ws_size,
                              hipStream_t stream) {
    const float* means = (const float*)d_in[0];
    const float* gt    = (const float*)d_in[1];
    const int B = 4;
    const int N = in_sizes[0] / (B * 3);   // 8192
    const int M = in_sizes[1] / (B * 3);   // 8192

    unsigned int* mins  = (unsigned int*)d_ws;
    unsigned int* d1min = mins;            // B*N entries
    unsigned int* d2min = mins + (size_t)B * N; // B*M entries

    const int total = B * N + B * M;
    cd_init<<<(total + 255) / 256, 256, 0, stream>>>(mins, total);

    const int blocks = B * (N / MEANS_PER_BLOCK) * SPLITS; // 256 blocks
    cd_main<<<blocks, 256, 0, stream>>>(means, gt, d1min, d2min, B, N, M);

    cd_final<<<1, 256, 0, stream>>>(d1min, d2min, (float*)d_out, B, N, M);
}